// FactorizedVectorQuantize_54778012893270
// MI455X (gfx1250) — compile-verified
//
#include <hip/hip_runtime.h>
#include <hip/hip_bf16.h>
#include <math.h>

// FactorizedVectorQuantize forward for MI455X (gfx1250, wave32).
// The 16384x8192 (K=8) nearest-code similarity search runs on
// V_WMMA_F32_16X16X4_F32 (exact f32, matches reference numerics).
// The codebook N-range is split 4-ways: each block stages a 2048-row slice
// (64 KB) of the normalized codebook in LDS so multiple workgroups co-reside
// per WGP (vs one at 256 KB), and a tiny merge pass combines the 4 partial
// argmax results with first-index tie-breaking.

typedef __attribute__((ext_vector_type(2))) float v2f;
typedef __attribute__((ext_vector_type(8))) float v8f;

#define B_    16
#define D_    1024
#define T_    1024
#define BT_   (B_ * T_)
#define CBS_  8192
#define CBD_  8
#define NSPLIT 4
#define ROWS_PER_SPLIT (CBS_ / NSPLIT)   // 2048 rows = 64 KB in LDS

// Workspace layout (float offsets). Total ~2.5 MB.
#define OFF_WIN    0          // 8*1024
#define OFF_WOUT   8192       // 1024*8
#define OFF_CBN    16384      // 8192*8 (L2-normalized codebook)
#define OFF_ZE     81920      // 16384*8 (raw z_e)
#define OFF_ENC    212992     // 16384*8 (normalized z_e)
#define OFF_ZQ     344064     // 16384*8 (gathered codebook[idx])
#define OFF_IDX    475136     // 16384 ints
#define OFF_PBEST  491520     // 4*16384 floats (per-split best similarity)
#define OFF_PIDX   557056     // 4*16384 ints   (per-split best index)

// ---------------------------------------------------------------------------
// Kernel 0: weight-norm W_in, W_out; L2-normalize codebook.
// blocks 0..7: W_in rows; 8..11: W_out rows; 12..43: codebook rows.
__global__ void fvq_prep(const float* __restrict__ in_v,
                         const float* __restrict__ in_g,
                         const float* __restrict__ out_v,
                         const float* __restrict__ out_g,
                         const float* __restrict__ cb,
                         float* __restrict__ ws) {
  __shared__ float red[256];
  __shared__ float s_norm;
  const int bid = blockIdx.x, tid = threadIdx.x;

  if (bid < 8) {
    // W_in row `bid`: norm over D=1024, then scale by g/norm.
    float s = 0.f;
    for (int j = tid; j < D_; j += 256) {
      float v = in_v[bid * D_ + j];
      s += v * v;
    }
    red[tid] = s;
    __syncthreads();
    for (int off = 128; off > 0; off >>= 1) {
      if (tid < off) red[tid] += red[tid + off];
      __syncthreads();
    }
    if (tid == 0) s_norm = sqrtf(red[0]);
    __syncthreads();
    const float scale = in_g[bid] / s_norm;
    for (int j = tid; j < D_; j += 256)
      ws[OFF_WIN + bid * D_ + j] = in_v[bid * D_ + j] * scale;
  } else if (bid < 12) {
    // W_out: one thread per output row (norm over CB_DIM=8).
    const int row = (bid - 8) * 256 + tid;  // 0..1023
    float v[8], s = 0.f;
#pragma unroll
    for (int c = 0; c < 8; ++c) { v[c] = out_v[row * 8 + c]; s += v[c] * v[c]; }
    const float scale = out_g[row] / sqrtf(s);
#pragma unroll
    for (int c = 0; c < 8; ++c) ws[OFF_WOUT + row * 8 + c] = v[c] * scale;
  } else {
    // Codebook L2-normalize (F.normalize semantics: /max(norm, 1e-12)).
    const int row = (bid - 12) * 256 + tid;  // 0..8191
    float v[8], s = 0.f;
#pragma unroll
    for (int c = 0; c < 8; ++c) { v[c] = cb[row * 8 + c]; s += v[c] * v[c]; }
    const float inv = 1.0f / fmaxf(sqrtf(s), 1e-12f);
#pragma unroll
    for (int c = 0; c < 8; ++c) ws[OFF_CBN + row * 8 + c] = v[c] * inv;
  }
}

// ---------------------------------------------------------------------------
// Kernel 1: z_e[b,t,c] = sum_d z[b,d,t] * W_in[c,d] + in_b[c]; also enc
// (normalized rows). Block: 256 threads = 8 d-groups x 32 t-lanes (coalesced
// over t), LDS reduce across d-groups. Grid: 16 b * 32 t-chunks = 512 blocks.
__global__ void fvq_ze(const float* __restrict__ z,
                       const float* __restrict__ in_b,
                       float* __restrict__ ws) {
  __shared__ float sWin[8 * 1024];   // 32 KB: full W_in
  __shared__ float sRed[256 * 8];    // 8 KB : per-thread partials
  __shared__ float sZe[32 * 8];      // 1 KB : finished rows

  const int tid = threadIdx.x;
  for (int i = tid; i < 8 * 1024; i += 256) sWin[i] = ws[OFF_WIN + i];
  __syncthreads();

  const int b  = blockIdx.x >> 5;
  const int t0 = (blockIdx.x & 31) * 32;
  const int g  = tid >> 5;          // d-group 0..7 (128 d's each)
  const int tl = tid & 31;          // t-lane
  const int t  = t0 + tl;

  const float* zp = z + (size_t)b * D_ * T_ + (size_t)(g * 128) * T_ + t;
  const float* wrow = &sWin[g * 128];

  float acc[8] = {0, 0, 0, 0, 0, 0, 0, 0};
#pragma unroll 4
  for (int dd = 0; dd < 128; ++dd) {
    const float v = zp[(size_t)dd * T_];   // lanes consecutive in t: coalesced
#pragma unroll
    for (int c = 0; c < 8; ++c) acc[c] += v * wrow[c * 1024 + dd];
  }
#pragma unroll
  for (int c = 0; c < 8; ++c) sRed[tid * 8 + c] = acc[c];
  __syncthreads();

  {  // reduce 8 d-groups: 256 threads = 32 t x 8 c
    const int t2 = tid >> 3, c2 = tid & 7;
    float s = in_b[c2];
#pragma unroll
    for (int gg = 0; gg < 8; ++gg) s += sRed[(gg * 32 + t2) * 8 + c2];
    sZe[t2 * 8 + c2] = s;
  }
  __syncthreads();

  if (tid < 32) {
    const int bt = b * T_ + t0 + tid;
    float r[8], n2 = 0.f;
#pragma unroll
    for (int c = 0; c < 8; ++c) { r[c] = sZe[tid * 8 + c]; n2 += r[c] * r[c]; }
    const float inv = 1.0f / fmaxf(sqrtf(n2), 1e-12f);
#pragma unroll
    for (int c = 0; c < 8; ++c) {
      ws[OFF_ZE  + bt * 8 + c] = r[c];
      ws[OFF_ENC + bt * 8 + c] = r[c] * inv;
    }
  }
}

// ---------------------------------------------------------------------------
// Kernel 2: partial argmax_n enc[m]·cbn[n] via V_WMMA_F32_16X16X4_F32.
// Grid: 128 M-groups x 4 codebook quarters = 512 blocks; block = 8 waves.
// Each block stages its 2048-row codebook quarter (64 KB) in LDS; each wave
// owns one 16-row M-tile and loops 128 N-tiles (K=8 as two chained K=4
// WMMAs, two independent accumulator chains via the x2 unroll).
// C layout: VGPR r -> (M = r + 8*(lane>=16), N = lane&15); running per-lane
// argmax then width-16 xor-shuffle reduce (first-index tie-break).
__global__ void fvq_argmax(float* __restrict__ ws) {
  extern __shared__ float scb[];  // 16384 floats = 64 KB

  const int tid = threadIdx.x;
  const int mgroup = blockIdx.x >> 2;
  const int q = blockIdx.x & 3;           // codebook quarter

  {  // cooperative load of this quarter, float4 granularity
    const float4* src = (const float4*)(ws + OFF_CBN + q * ROWS_PER_SPLIT * 8);
    float4* dst = (float4*)scb;
    for (int i = tid; i < ROWS_PER_SPLIT * 2; i += 256) dst[i] = src[i];
  }
  __syncthreads();

  const int wave = tid >> 5, lane = tid & 31;
  const int hf = lane >> 4, l15 = lane & 15;
  const int mtile = mgroup * 8 + wave;

  // A tile (16x8 f32): lane holds K pair (2*hf, 2*hf+1) of row m, and +4 pair.
  const float* enc = ws + OFF_ENC;
  const int m = mtile * 16 + l15;
  const v2f a01 = *(const v2f*)&enc[m * 8 + hf * 2];
  const v2f a23 = *(const v2f*)&enc[m * 8 + 4 + hf * 2];

  float best[8];
  int bidx[8];
#pragma unroll
  for (int r = 0; r < 8; ++r) { best[r] = -3.4e38f; bidx[r] = 0; }

  // Prime the pipeline with tile 0 of this quarter.
  const int col0 = l15 * 8 + hf * 2;      // lane's base within a 16-row tile
  v2f b01 = *(const v2f*)&scb[col0];
  v2f b23 = *(const v2f*)&scb[col0 + 4];

#pragma unroll 2
  for (int nt = 0; nt < ROWS_PER_SPLIT / 16; ++nt) {
    const int n = q * ROWS_PER_SPLIT + nt * 16 + l15;  // global codebook index
    // Prefetch next tile (branch-free wrap keeps EXEC all-ones for WMMA).
    const int pf = ((nt + 1) & (ROWS_PER_SPLIT / 16 - 1)) * 128 + col0;
    const v2f nb01 = *(const v2f*)&scb[pf];
    const v2f nb23 = *(const v2f*)&scb[pf + 4];

    v8f c = {};
    // args: (neg_a, A, neg_b, B, c_mod, C, reuse_a, reuse_b)
    c = __builtin_amdgcn_wmma_f32_16x16x4_f32(false, a01, false, b01,
                                              (short)0, c, false, false);
    c = __builtin_amdgcn_wmma_f32_16x16x4_f32(false, a23, false, b23,
                                              (short)0, c, false, false);
#pragma unroll
    for (int r = 0; r < 8; ++r) {
      const float v = c[r];
      if (v > best[r]) { best[r] = v; bidx[r] = n; }  // '>' keeps first max
    }
    b01 = nb01;
    b23 = nb23;
  }

  // Combine the 16 lanes holding the same 8 rows (lanes 0-15: M=r, 16-31: M=8+r)
#pragma unroll
  for (int r = 0; r < 8; ++r) {
    float v = best[r];
    int ix = bidx[r];
    for (int off = 8; off >= 1; off >>= 1) {
      const float ov = __shfl_xor(v, off, 16);
      const int   oi = __shfl_xor(ix, off, 16);
      if (ov > v || (ov == v && oi < ix)) { v = ov; ix = oi; }
    }
    best[r] = v;
    bidx[r] = ix;
  }

  if (l15 == 0) {
    float* pbest = ws + OFF_PBEST;
    int* pidx = (int*)(ws + OFF_PIDX);
    const int mbase = mtile * 16 + hf * 8;
#pragma unroll
    for (int r = 0; r < 8; ++r) {
      pbest[q * BT_ + mbase + r] = best[r];
      pidx [q * BT_ + mbase + r] = bidx[r];
    }
  }
}

// ---------------------------------------------------------------------------
// Kernel 2b: merge the 4 per-quarter partials. Ascending quarter scan with
// strict '>' preserves jnp.argmax first-occurrence tie-breaking (lower
// quarters contain strictly lower codebook indices).
__global__ void fvq_merge(float* __restrict__ ws, float* __restrict__ d_out) {
  const int m = blockIdx.x * 256 + threadIdx.x;   // 0..16383
  const float* pbest = ws + OFF_PBEST;
  const int* pidx = (const int*)(ws + OFF_PIDX);

  float bv = pbest[m];
  int bi = pidx[m];
#pragma unroll
  for (int qq = 1; qq < NSPLIT; ++qq) {
    const float v = pbest[qq * BT_ + m];
    const int i = pidx[qq * BT_ + m];
    if (v > bv) { bv = v; bi = i; }
  }
  ((int*)(ws + OFF_IDX))[m] = bi;
  d_out[(size_t)B_ * D_ * T_ + m] = (float)bi;    // indices (exact in f32)
}

// ---------------------------------------------------------------------------
// Kernel 3: gather z_q = codebook[idx] (raw) and commit_loss[b] =
// 1.25 * mean((z_e - z_q)^2) over (T, CB_DIM). One block per batch element.
__global__ void fvq_loss(const float* __restrict__ cb,
                         float* __restrict__ ws,
                         float* __restrict__ d_out) {
  const int b = blockIdx.x, tid = threadIdx.x;
  const int* idx = (const int*)(ws + OFF_IDX);
  const float* ze = ws + OFF_ZE;
  float* zq = ws + OFF_ZQ;

  float s = 0.f;
  for (int k = 0; k < 4; ++k) {
    const int bt = b * T_ + tid + k * 256;
    const int id = idx[bt];
    const float* q = cb + (size_t)id * 8;
#pragma unroll
    for (int c = 0; c < 8; ++c) {
      const float qc = q[c];
      const float d = ze[bt * 8 + c] - qc;
      s += d * d;
      zq[bt * 8 + c] = qc;  // STE forward value == raw codebook row
    }
  }
  __shared__ float red[256];
  red[tid] = s;
  __syncthreads();
  for (int off = 128; off > 0; off >>= 1) {
    if (tid < off) red[tid] += red[tid + off];
    __syncthreads();
  }
  if (tid == 0)
    d_out[(size_t)B_ * D_ * T_ + BT_ + b] = 1.25f * red[0] / (float)(T_ * CBD_);
}

// ---------------------------------------------------------------------------
// Kernel 4: out[b,d,t] = dot(z_q[b,t,:], W_out[d,:]) + out_b[d].
// One thread per output element; lanes along t (coalesced 64 MB write).
__global__ void fvq_out(const float* __restrict__ ws,
                        const float* __restrict__ out_b,
                        float* __restrict__ d_out) {
  const size_t gid = (size_t)blockIdx.x * 256 + threadIdx.x;
  const int t = (int)(gid & (T_ - 1));
  const int d = (int)((gid >> 10) & (D_ - 1));
  const int b = (int)(gid >> 20);
  const int bt = b * T_ + t;

  const float4* zq4 = (const float4*)(ws + OFF_ZQ);
  const float4* wo4 = (const float4*)(ws + OFF_WOUT);
  const float4 q0 = zq4[bt * 2], q1 = zq4[bt * 2 + 1];
  const float4 w0 = wo4[d * 2],  w1 = wo4[d * 2 + 1];

  float acc = out_b[d];
  acc += q0.x * w0.x + q0.y * w0.y + q0.z * w0.z + q0.w * w0.w;
  acc += q1.x * w1.x + q1.y * w1.y + q1.z * w1.z + q1.w * w1.w;
  d_out[gid] = acc;
}

// ---------------------------------------------------------------------------
extern "C" void kernel_launch(void* const* d_in, const int* in_sizes, int n_in,
                              void* d_out, int out_size, void* d_ws,
                              size_t ws_size, hipStream_t stream) {
  (void)in_sizes; (void)n_in; (void)out_size; (void)ws_size;
  const float* z     = (const float*)d_in[0];
  const float* in_v  = (const float*)d_in[1];
  const float* in_g  = (const float*)d_in[2];
  const float* in_b  = (const float*)d_in[3];
  const float* cb    = (const float*)d_in[4];
  const float* out_v = (const float*)d_in[5];
  const float* out_g = (const float*)d_in[6];
  const float* out_b = (const float*)d_in[7];
  float* out = (float*)d_out;
  float* ws  = (float*)d_ws;

  fvq_prep<<<44, 256, 0, stream>>>(in_v, in_g, out_v, out_g, cb, ws);
  fvq_ze<<<512, 256, 0, stream>>>(z, in_b, ws);
  fvq_argmax<<<128 * NSPLIT, 256, ROWS_PER_SPLIT * CBD_ * sizeof(float),
               stream>>>(ws);
  fvq_merge<<<BT_ / 256, 256, 0, stream>>>(ws, out);
  fvq_loss<<<16, 256, 0, stream>>>(cb, ws, out);
  fvq_out<<<(B_ * D_ * T_) / 256, 256, 0, stream>>>(ws, out_b, out);
}